// rpnProposalLayer_76527727280563
// MI455X (gfx1250) — compile-verified
//
#include <hip/hip_runtime.h>
#include <hip/hip_bf16.h>
#include <stdint.h>

#define BATCH     16
#define POOL      28128          // 6000 + 6000 + 12288 + 3072 + 768
#define POST_N    300
#define PRE_TOPN  6000
#define NMS_TH    0.7f
#define NEG_SENT  (-3.0e38f)

// Base anchor w/h per (level, ratio) BEFORE the x8 scale, matching numpy's
// banker's rounding in _base_anchors (verified by hand for sb=4,8,16,32,64).
__constant__ int d_AW[5][3] = {{6,4,3},{11,8,6},{23,16,11},{45,32,23},{91,64,45}};
__constant__ int d_AH[5][3] = {{3,4,6},{6,8,12},{12,16,22},{22,32,46},{46,64,90}};

__device__ __forceinline__ unsigned keyOf(float f) {
    unsigned u = __float_as_uint(f);
    return (u & 0x80000000u) ? ~u : (u | 0x80000000u);
}

__device__ __forceinline__ float4 decode_clip(int l, int stride, int Wg, int n_idx,
                                              const float* __restrict__ reg,
                                              float im_h, float im_w) {
    int a    = n_idx % 3;
    int cell = n_idx / 3;
    int x    = cell % Wg;
    int y    = cell / Wg;
    float sw = (float)d_AW[l][a] * 8.0f;       // anchor width  (x2-x1+1)
    float sh = (float)d_AH[l][a] * 8.0f;       // anchor height
    float cx = (float)x * (float)stride + 0.5f * (float)stride;  // = shift + (sb-1)/2 + 0.5
    float cy = (float)y * (float)stride + 0.5f * (float)stride;
    const float* d = reg + (size_t)n_idx * 4;
    float pcx = d[0] * sw + cx;
    float pcy = d[1] * sh + cy;
    float pw  = expf(d[2]) * sw;
    float ph  = expf(d[3]) * sh;
    float4 b;
    b.x = fminf(fmaxf(pcx - 0.5f * pw, 0.0f), im_w - 1.0f);
    b.y = fminf(fmaxf(pcy - 0.5f * ph, 0.0f), im_h - 1.0f);
    b.z = fminf(fmaxf(pcx + 0.5f * pw, 0.0f), im_w - 1.0f);
    b.w = fminf(fmaxf(pcy + 0.5f * ph, 0.0f), im_h - 1.0f);
    return b;
}

// ---------------- init: zero histograms + counters -------------------------
__global__ void k_init(unsigned* __restrict__ p, int n) {
    int i = blockIdx.x * 256 + threadIdx.x;
    if (i < n) p[i] = 0;
}

// ---------------- pass 1: 12-bit histogram of score keys -------------------
__global__ void k_hist1(const float* __restrict__ probs, int n, int l,
                        unsigned* __restrict__ hist1) {
    __shared__ unsigned h[4096];
    for (int i = threadIdx.x; i < 4096; i += 256) h[i] = 0;
    __syncthreads();
    int b = blockIdx.y;
    const float* pb = probs + (size_t)b * n * 2;
    int base = blockIdx.x * 4096;
    for (int it = 0; it < 16; ++it) {
        int i = base + it * 256 + threadIdx.x;
        if (i < n) atomicAdd(&h[keyOf(pb[(size_t)i * 2 + 1]) >> 20], 1u);
    }
    __syncthreads();
    unsigned* g = hist1 + ((size_t)l * BATCH + b) * 4096;
    for (int i = threadIdx.x; i < 4096; i += 256)
        if (h[i]) atomicAdd(&g[i], h[i]);
}

__global__ void k_thresh1(const unsigned* __restrict__ hist1,
                          unsigned* __restrict__ thr1, unsigned* __restrict__ abv1) {
    if (threadIdx.x != 0) return;
    int id = blockIdx.x;                       // l*16 + b
    const unsigned* h = hist1 + (size_t)id * 4096;
    unsigned acc = 0; int bin;
    for (bin = 4095; bin >= 0; --bin) {
        unsigned c = h[bin];
        if (acc + c >= PRE_TOPN) break;
        acc += c;
    }
    if (bin < 0) bin = 0;
    thr1[id] = (unsigned)bin; abv1[id] = acc;
}

// ---------------- pass 2: refine boundary bin with next 12 bits ------------
__global__ void k_hist2(const float* __restrict__ probs, int n, int l,
                        const unsigned* __restrict__ thr1,
                        unsigned* __restrict__ hist2) {
    __shared__ unsigned h[4096];
    for (int i = threadIdx.x; i < 4096; i += 256) h[i] = 0;
    __syncthreads();
    int b = blockIdx.y;
    unsigned t1 = thr1[l * BATCH + b];
    const float* pb = probs + (size_t)b * n * 2;
    int base = blockIdx.x * 4096;
    for (int it = 0; it < 16; ++it) {
        int i = base + it * 256 + threadIdx.x;
        if (i < n) {
            unsigned k = keyOf(pb[(size_t)i * 2 + 1]);
            if ((k >> 20) == t1) atomicAdd(&h[(k >> 8) & 0xFFFu], 1u);
        }
    }
    __syncthreads();
    unsigned* g = hist2 + ((size_t)l * BATCH + b) * 4096;
    for (int i = threadIdx.x; i < 4096; i += 256)
        if (h[i]) atomicAdd(&g[i], h[i]);
}

__global__ void k_thresh2(const unsigned* __restrict__ hist2,
                          const unsigned* __restrict__ abv1,
                          unsigned* __restrict__ thr2, unsigned* __restrict__ quo) {
    if (threadIdx.x != 0) return;
    int id = blockIdx.x;
    unsigned need = PRE_TOPN - abv1[id];
    const unsigned* h = hist2 + (size_t)id * 4096;
    unsigned acc = 0; int bin;
    for (bin = 4095; bin >= 0; --bin) {
        unsigned c = h[bin];
        if (acc + c >= need) break;
        acc += c;
    }
    if (bin < 0) bin = 0;
    thr2[id] = (unsigned)bin; quo[id] = need - acc;
}

// ---------------- compact survivors of levels 0/1 + fused decode -----------
__global__ void k_compact(const float* __restrict__ probs, const float* __restrict__ regs,
                          const float* __restrict__ img_info,
                          int n, int l, int stride, int Wg, int pool_base,
                          const unsigned* __restrict__ thr1, const unsigned* __restrict__ thr2,
                          const unsigned* __restrict__ quo,
                          unsigned* __restrict__ cnt, unsigned* __restrict__ bcnt,
                          float* __restrict__ p_sc, float4* __restrict__ p_box) {
    int b  = blockIdx.y;
    int id = l * BATCH + b;
    unsigned t1 = thr1[id], t2 = thr2[id], q = quo[id];
    float im_h = img_info[b * 3 + 0], im_w = img_info[b * 3 + 1];
    const float* pb = probs + (size_t)b * n * 2;
    const float* rb = regs  + (size_t)b * n * 4;
    int base = blockIdx.x * 4096;
    for (int it = 0; it < 16; ++it) {
        int i = base + it * 256 + threadIdx.x;
        if (i >= n) continue;
        float s = pb[(size_t)i * 2 + 1];
        unsigned k  = keyOf(s);
        unsigned b1 = k >> 20;
        bool take = false;
        if (b1 > t1) take = true;
        else if (b1 == t1) {
            unsigned b2 = (k >> 8) & 0xFFFu;
            if (b2 > t2)       take = true;
            else if (b2 == t2) take = (atomicAdd(&bcnt[id], 1u) < q);
        }
        if (take) {
            unsigned slot = atomicAdd(&cnt[id], 1u);
            float4 box = decode_clip(l, stride, Wg, i, rb, im_h, im_w);
            size_t o = (size_t)b * POOL + pool_base + slot;
            p_sc[o]  = s;
            p_box[o] = box;
        }
    }
}

// ---------------- levels 2..4: all anchors pass through --------------------
__global__ void k_decode_rest(const float* __restrict__ p2, const float* __restrict__ p3,
                              const float* __restrict__ p4,
                              const float* __restrict__ r2, const float* __restrict__ r3,
                              const float* __restrict__ r4,
                              const float* __restrict__ img_info,
                              float* __restrict__ p_sc, float4* __restrict__ p_box) {
    int b   = blockIdx.y;
    int idx = blockIdx.x * 256 + threadIdx.x;   // [0, 16128)
    int l, local, pbase, n, stride, Wg;
    const float *pp, *rr;
    if (idx < 12288)      { l=2; local=idx;         pbase=12000; n=12288; stride=16; Wg=64; pp=p2; rr=r2; }
    else if (idx < 15360) { l=3; local=idx-12288;   pbase=24288; n=3072;  stride=32; Wg=32; pp=p3; rr=r3; }
    else                  { l=4; local=idx-15360;   pbase=27360; n=768;   stride=64; Wg=16; pp=p4; rr=r4; }
    float im_h = img_info[b * 3 + 0], im_w = img_info[b * 3 + 1];
    float s = pp[((size_t)b * n + local) * 2 + 1];
    float4 box = decode_clip(l, stride, Wg, local, rr + (size_t)b * n * 4, im_h, im_w);
    size_t o = (size_t)b * POOL + pbase + local;
    p_sc[o]  = s;
    p_box[o] = box;
}

// ---------------- NMS: argmax-based (sort-free), scores LDS-resident -------
typedef unsigned int tdm_v4u __attribute__((ext_vector_type(4)));
typedef int          tdm_v8i __attribute__((ext_vector_type(8)));
typedef int          tdm_v4i __attribute__((ext_vector_type(4)));

__global__ void __launch_bounds__(1024)
k_nms(const float* __restrict__ p_sc, const float4* __restrict__ p_box,
      float* __restrict__ out) {
    extern __shared__ float smem[];
    float* s_sc  = smem;                         // POOL scores @ LDS offset 0 (TDM dest)
    float* s_rv  = smem + POOL;                  // 1024 reduction values
    int*   s_ri  = (int*)(smem + POOL + 1024);   // 1024 reduction indices
    float* s_box = smem + POOL + 2048;           // x1,y1,x2,y2,area,flag

    int b = blockIdx.x;
    const float4* bx = p_box + (size_t)b * POOL;

    // Stage this image's 112.5 KB score vector into LDS via the Tensor Data
    // Mover: 1-D D# (count=1, data_size=4B, tensor_dim0=tile_dim0=POOL).
    // amdgpu-toolchain clang-23 form: 6 args (v4u, v8i, v4i, v4i, v8i, cpol).
    if (threadIdx.x == 0) {
        unsigned long long ga = (unsigned long long)(const void*)(p_sc + (size_t)b * POOL);
        tdm_v4u g0;
        g0[0] = 1u;                                            // count=1, user descriptor
        g0[1] = 0u;                                            // lds_addr = 0
        g0[2] = (unsigned)(ga & 0xFFFFFFFFull);                // global_addr[31:0]
        g0[3] = (unsigned)((ga >> 32) & 0x01FFFFFFull) | (2u << 30); // addr[56:32] | type=2
        tdm_v8i g1;
        g1[0] = (int)(2u << 16);                               // data_size=4B
        g1[1] = (int)(((unsigned)POOL & 0xFFFFu) << 16);       // tensor_dim0[15:0]
        g1[2] = (int)((((unsigned)POOL >> 16) & 0xFFFFu) | (1u << 16)); // dim0 hi | tensor_dim1=1
        g1[3] = (int)((unsigned)POOL << 16);                   // tile_dim0 = POOL
        g1[4] = 1;                                             // tile_dim1=1, tile_dim2=0
        g1[5] = POOL;                                          // tensor_dim0_stride lo
        g1[6] = 0;
        g1[7] = 0;
        tdm_v4i g2 = {0, 0, 0, 0};
        tdm_v4i g3 = {0, 0, 0, 0};
        tdm_v8i g4 = {0, 0, 0, 0, 0, 0, 0, 0};
        __builtin_amdgcn_tensor_load_to_lds(g0, g1, g2, g3, g4, 0);
        __builtin_amdgcn_s_wait_tensorcnt(0);
    }
    __syncthreads();

    float* outb = out + (size_t)b * POST_N * 5;

    for (int k = 0; k < POST_N; ++k) {
        // ---- parallel argmax over LDS scores (smaller index wins ties) ----
        float bv = NEG_SENT; int bi = -1;
        for (int i = threadIdx.x; i < POOL; i += 1024) {
            float v = s_sc[i];
            if (v > bv) { bv = v; bi = i; }
        }
        s_rv[threadIdx.x] = bv; s_ri[threadIdx.x] = bi;
        __syncthreads();
        for (int off = 512; off > 0; off >>= 1) {
            if (threadIdx.x < off) {
                float v1 = s_rv[threadIdx.x],       v2 = s_rv[threadIdx.x + off];
                int   i1 = s_ri[threadIdx.x],       i2 = s_ri[threadIdx.x + off];
                if (v2 > v1 || (v2 == v1 && i2 >= 0 && (i1 < 0 || i2 < i1))) {
                    s_rv[threadIdx.x] = v2; s_ri[threadIdx.x] = i2;
                }
            }
            __syncthreads();
        }
        if (threadIdx.x == 0) {
            int pick = s_ri[0];
            if (pick < 0) {
                s_box[5] = 0.0f;
            } else {
                float4 bb = bx[pick];
                s_box[0] = bb.x; s_box[1] = bb.y; s_box[2] = bb.z; s_box[3] = bb.w;
                s_box[4] = (bb.z - bb.x + 1.0f) * (bb.w - bb.y + 1.0f);
                s_box[5] = 1.0f;
                s_sc[pick] = NEG_SENT;                       // self-suppress
                outb[k * 5 + 0] = (float)b;
                outb[k * 5 + 1] = bb.x; outb[k * 5 + 2] = bb.y;
                outb[k * 5 + 3] = bb.z; outb[k * 5 + 4] = bb.w;
            }
        }
        __syncthreads();
        if (s_box[5] == 0.0f) {                              // exhausted: zero-fill rest
            for (int kk = k + (int)threadIdx.x; kk < POST_N; kk += 1024) {
                outb[kk * 5 + 0] = (float)b;
                outb[kk * 5 + 1] = 0.0f; outb[kk * 5 + 2] = 0.0f;
                outb[kk * 5 + 3] = 0.0f; outb[kk * 5 + 4] = 0.0f;
            }
            break;
        }
        if (k == POST_N - 1) break;
        // ---- suppression pass: IoU vs picked box over live candidates -----
        float x1 = s_box[0], y1 = s_box[1], x2 = s_box[2], y2 = s_box[3], ar = s_box[4];
        for (int i = threadIdx.x; i < POOL; i += 1024) {
            float v = s_sc[i];
            if (v == NEG_SENT) continue;
            if (i + 1024 < POOL) __builtin_prefetch(&bx[i + 1024], 0, 0);
            float4 bb = bx[i];
            float xx1 = fmaxf(x1, bb.x), yy1 = fmaxf(y1, bb.y);
            float xx2 = fminf(x2, bb.z), yy2 = fminf(y2, bb.w);
            float iw = fmaxf(xx2 - xx1 + 1.0f, 0.0f);
            float ih = fmaxf(yy2 - yy1 + 1.0f, 0.0f);
            float inter = iw * ih;
            float a2 = (bb.z - bb.x + 1.0f) * (bb.w - bb.y + 1.0f);
            float iou = inter / (ar + a2 - inter);
            if (iou > NMS_TH) s_sc[i] = NEG_SENT;
        }
        __syncthreads();
    }
}

// ============================ host side ====================================
extern "C" void kernel_launch(void* const* d_in, const int* in_sizes, int n_in,
                              void* d_out, int out_size, void* d_ws, size_t ws_size,
                              hipStream_t stream) {
    // setup_inputs() dict order: probs0,xreg0,probs1,xreg1,...,probs4,xreg4,img_info
    const float* probs0 = (const float*)d_in[0];
    const float* xreg0  = (const float*)d_in[1];
    const float* probs1 = (const float*)d_in[2];
    const float* xreg1  = (const float*)d_in[3];
    const float* probs2 = (const float*)d_in[4];
    const float* xreg2  = (const float*)d_in[5];
    const float* probs3 = (const float*)d_in[6];
    const float* xreg3  = (const float*)d_in[7];
    const float* probs4 = (const float*)d_in[8];
    const float* xreg4  = (const float*)d_in[9];
    const float* info   = (const float*)d_in[10];
    float* out = (float*)d_out;

    // workspace layout
    char* base = (char*)d_ws;
    float*    p_sc  = (float*)base;                                 // BATCH*POOL f32
    float4*   p_box = (float4*)(base + (size_t)BATCH * POOL * 4);   // BATCH*POOL f32x4
    unsigned* hist1 = (unsigned*)(base + (size_t)BATCH * POOL * 4 + (size_t)BATCH * POOL * 16);
    unsigned* hist2 = hist1 + 2 * BATCH * 4096;
    unsigned* thr1  = hist2 + 2 * BATCH * 4096;
    unsigned* abv1  = thr1 + 2 * BATCH;
    unsigned* thr2  = abv1 + 2 * BATCH;
    unsigned* quo   = thr2 + 2 * BATCH;
    unsigned* cnt   = quo  + 2 * BATCH;
    unsigned* bcnt  = cnt  + 2 * BATCH;

    const int n0 = 196608, n1 = 49152;
    const int zeroN = 2 * 2 * BATCH * 4096 + 6 * 2 * BATCH;  // hists + 6 small arrays

    k_init<<<dim3((zeroN + 255) / 256), dim3(256), 0, stream>>>(hist1, zeroN);

    k_hist1<<<dim3(48, BATCH), dim3(256), 0, stream>>>(probs0, n0, 0, hist1);
    k_hist1<<<dim3(12, BATCH), dim3(256), 0, stream>>>(probs1, n1, 1, hist1);
    k_thresh1<<<dim3(2 * BATCH), dim3(32), 0, stream>>>(hist1, thr1, abv1);

    k_hist2<<<dim3(48, BATCH), dim3(256), 0, stream>>>(probs0, n0, 0, thr1, hist2);
    k_hist2<<<dim3(12, BATCH), dim3(256), 0, stream>>>(probs1, n1, 1, thr1, hist2);
    k_thresh2<<<dim3(2 * BATCH), dim3(32), 0, stream>>>(hist2, abv1, thr2, quo);

    k_compact<<<dim3(48, BATCH), dim3(256), 0, stream>>>(probs0, xreg0, info,
            n0, 0, 4, 256, 0,     thr1, thr2, quo, cnt, bcnt, p_sc, p_box);
    k_compact<<<dim3(12, BATCH), dim3(256), 0, stream>>>(probs1, xreg1, info,
            n1, 1, 8, 128, 6000,  thr1, thr2, quo, cnt, bcnt, p_sc, p_box);

    k_decode_rest<<<dim3(63, BATCH), dim3(256), 0, stream>>>(probs2, probs3, probs4,
            xreg2, xreg3, xreg4, info, p_sc, p_box);

    size_t smem = (size_t)(POOL + 1024 + 1024 + 8) * sizeof(float);  // ~120.8 KB (< 320 KB WGP LDS)
    k_nms<<<dim3(BATCH), dim3(1024), smem, stream>>>(p_sc, p_box, out);
}